// GNNModel_67697274520408
// MI455X (gfx1250) — compile-verified
//
#include <hip/hip_runtime.h>
#include <hip/hip_bf16.h>

typedef __attribute__((ext_vector_type(2))) float v2f;
typedef __attribute__((ext_vector_type(8))) float v8f;

// ---------------------------------------------------------------------------
// GEMM C[M,N] = A[M,K] * B[K,N], all row-major f32, via V_WMMA_F32_16X16X4_F32.
// One wave32 per 16x16 output tile; K stepped by 4.
// A 16x4 layout: lane 0-15 -> M=lane, VGPR0=K0,VGPR1=K1; lanes 16-31 -> K2,K3.
// B 4x16 layout mirrored; C/D: VGPR r -> M=r (lanes 0-15) / M=8+r (lanes 16-31).
// ---------------------------------------------------------------------------
__global__ void gemm_wmma_f32(const float* __restrict__ A,
                              const float* __restrict__ B,
                              float* __restrict__ C,
                              int M, int N, int K) {
  const int lane   = threadIdx.x & 31;
  const int waveId = blockIdx.x * (blockDim.x >> 5) + (threadIdx.x >> 5);
  const int tilesN = N >> 4;
  const int tilesM = (M + 15) >> 4;
  if (waveId >= tilesM * tilesN) return;   // wave-uniform exit; EXEC all-1s below
  const int tm = waveId / tilesN;
  const int tn = waveId - tm * tilesN;

  int row = tm * 16 + (lane & 15);
  if (row > M - 1) row = M - 1;            // clamp (keeps EXEC full for WMMA)
  const int col = tn * 16 + (lane & 15);
  const int kh  = (lane >> 4) << 1;        // 0 for lanes 0-15, 2 for 16-31

  v8f acc = {};
  for (int k0 = 0; k0 < K; k0 += 4) {
    v2f a, b;
    a.x = A[(size_t)row * K + k0 + kh];
    a.y = A[(size_t)row * K + k0 + kh + 1];
    b.x = B[(size_t)(k0 + kh) * N + col];
    b.y = B[(size_t)(k0 + kh + 1) * N + col];
    acc = __builtin_amdgcn_wmma_f32_16x16x4_f32(false, a, false, b,
                                                (short)0, acc, false, false);
  }

  const int mbase = tm * 16 + ((lane >> 4) << 3);
#pragma unroll
  for (int r = 0; r < 8; ++r) {
    const int m = mbase + r;
    if (m < M) C[(size_t)m * N + col] = acc[r];
  }
}

// ---------------------------------------------------------------------------
// Degree count over dst (float atomics; counts are small integers, exact).
// ---------------------------------------------------------------------------
__global__ void degree_kernel(const long long* __restrict__ dst,
                              float* __restrict__ cnt, long long E) {
  long long e = (long long)blockIdx.x * blockDim.x + threadIdx.x;
  if (e < E) atomicAdd(&cnt[dst[e]], 1.0f);
}

// dinv = rsqrt(deg) with self-loop: deg = cnt + 1 (always >= 1)
__global__ void dinv_kernel(const float* __restrict__ cnt,
                            float* __restrict__ dinv, int Nn) {
  int i = blockIdx.x * blockDim.x + threadIdx.x;
  if (i < Nn) dinv[i] = rsqrtf(cnt[i] + 1.0f);
}

// ---------------------------------------------------------------------------
// Edge scatter: one wave32 per edge, 2 features per lane.
// agg[dst] += h[src] * dinv[src]*dinv[dst]   (64 features / edge)
// Tables are L2-resident (25.6 MB each, 192 MB L2).
// ---------------------------------------------------------------------------
__global__ void scatter_edges(const long long* __restrict__ src,
                              const long long* __restrict__ dst,
                              const float* __restrict__ h,
                              const float* __restrict__ dinv,
                              float* __restrict__ agg, long long E) {
  long long t = (long long)blockIdx.x * blockDim.x + threadIdx.x;
  long long e = t >> 5;
  if (e >= E) return;
  const int f = (int)(t & 31) * 2;
  const long long s = src[e];
  const long long d = dst[e];
  const float norm = dinv[s] * dinv[d];
  const float2 hv = *(const float2*)(h + (size_t)s * 64 + f);
  atomicAdd(agg + (size_t)d * 64 + f,     hv.x * norm);
  atomicAdd(agg + (size_t)d * 64 + f + 1, hv.y * norm);
}

// ---------------------------------------------------------------------------
// Finalize (in-place on agg): out = relu(agg + h*dinv^2 (self-loop) + bias)
// ---------------------------------------------------------------------------
__global__ void finalize_kernel(float* __restrict__ agg,
                                const float* __restrict__ h,
                                const float* __restrict__ dinv,
                                const float* __restrict__ bias, int Nn) {
  long long t = (long long)blockIdx.x * blockDim.x + threadIdx.x;
  long long i = t >> 6;
  int f = (int)(t & 63);
  if (i >= Nn) return;
  const float di = dinv[i];
  const size_t idx = (size_t)i * 64 + f;
  float v = agg[idx] + h[idx] * di * di + bias[f];
  agg[idx] = fmaxf(v, 0.0f);
}

// ---------------------------------------------------------------------------
// Per-feature column-sum with LDS reduction, then 64 atomics per block.
// ---------------------------------------------------------------------------
__global__ void reduce_mean_kernel(const float* __restrict__ a,
                                   float* __restrict__ gsum, int Nn) {
  __shared__ float sh[64];
  const int tid = threadIdx.x;
  if (tid < 64) sh[tid] = 0.0f;
  __syncthreads();
  const int f      = tid & 63;
  const int rowInB = tid >> 6;                 // 0..3 with 256 threads
  float acc = 0.0f;
  for (int i = blockIdx.x * 4 + rowInB; i < Nn; i += gridDim.x * 4)
    acc += a[(size_t)i * 64 + f];
  atomicAdd(&sh[f], acc);
  __syncthreads();
  if (tid < 64) atomicAdd(&gsum[tid], sh[tid]);
}

// out[0] = (gsum/N) @ Wfc + bfc
__global__ void final_fc_kernel(const float* __restrict__ gsum,
                                const float* __restrict__ Wfc,
                                const float* __restrict__ bfc,
                                float* __restrict__ out, float invN) {
  if (threadIdx.x == 0 && blockIdx.x == 0) {
    float s = 0.0f;
    for (int f = 0; f < 64; ++f) s += gsum[f] * invN * Wfc[f];
    out[0] = s + bfc[0];
  }
}

// ---------------------------------------------------------------------------
extern "C" void kernel_launch(void* const* d_in, const int* in_sizes, int n_in,
                              void* d_out, int out_size, void* d_ws, size_t ws_size,
                              hipStream_t stream) {
  const float*     x    = (const float*)d_in[0];
  const long long* edge = (const long long*)d_in[1];   // int64 [2, E]
  const float*     W1   = (const float*)d_in[2];
  const float*     b1   = (const float*)d_in[3];
  const float*     W2   = (const float*)d_in[4];
  const float*     b2   = (const float*)d_in[5];
  const float*     Wfc  = (const float*)d_in[6];
  const float*     bfc  = (const float*)d_in[7];
  float*           out  = (float*)d_out;

  const int       Nn = in_sizes[0] / 128;       // 100000 nodes
  const long long E  = (long long)in_sizes[1] / 2;
  const long long* srcIdx = edge;
  const long long* dstIdx = edge + E;

  // Workspace layout (floats): cnt[N], dinv[N], h[N*64], agg[N*64], gsum[64]
  float* cnt  = (float*)d_ws;
  float* dinv = cnt  + Nn;
  float* h    = dinv + Nn;
  float* agg  = h    + (size_t)Nn * 64;
  float* gsum = agg  + (size_t)Nn * 64;

  const int B = 256;
  const int degBlocks  = (int)((E + B - 1) / B);
  const int nodeBlocks = (Nn + B - 1) / B;
  const int scatBlocks = (int)((E * 32 + B - 1) / B);
  const int finBlocks  = (int)(((long long)Nn * 64 + B - 1) / B);
  const int tiles      = ((Nn + 15) / 16) * (64 / 16);
  const int gemmBlocks = (tiles + (B / 32) - 1) / (B / 32);

  // --- degrees / normalization ---
  hipMemsetAsync(cnt, 0, (size_t)Nn * sizeof(float), stream);
  degree_kernel<<<degBlocks, B, 0, stream>>>(dstIdx, cnt, E);
  dinv_kernel<<<nodeBlocks, B, 0, stream>>>(cnt, dinv, Nn);

  // --- layer 1: h1 = x @ W1 (f32 WMMA), aggregate, relu ---
  gemm_wmma_f32<<<gemmBlocks, B, 0, stream>>>(x, W1, h, Nn, 64, 128);
  hipMemsetAsync(agg, 0, (size_t)Nn * 64 * sizeof(float), stream);
  scatter_edges<<<scatBlocks, B, 0, stream>>>(srcIdx, dstIdx, h, dinv, agg, E);
  finalize_kernel<<<finBlocks, B, 0, stream>>>(agg, h, dinv, b1, Nn);

  // --- layer 2: h2 = out1 @ W2, aggregate, relu (buffers reused) ---
  gemm_wmma_f32<<<gemmBlocks, B, 0, stream>>>(agg, W2, h, Nn, 64, 64);
  hipMemsetAsync(agg, 0, (size_t)Nn * 64 * sizeof(float), stream);
  scatter_edges<<<scatBlocks, B, 0, stream>>>(srcIdx, dstIdx, h, dinv, agg, E);
  finalize_kernel<<<finBlocks, B, 0, stream>>>(agg, h, dinv, b2, Nn);

  // --- global mean pool + FC ---
  hipMemsetAsync(gsum, 0, 64 * sizeof(float), stream);
  reduce_mean_kernel<<<512, B, 0, stream>>>(agg, gsum, Nn);
  final_fc_kernel<<<1, 32, 0, stream>>>(gsum, Wfc, bfc, out, 1.0f / (float)Nn);
}